// MPG_65979287601499
// MI455X (gfx1250) — compile-verified
//
#include <hip/hip_runtime.h>
#include <hip/hip_bf16.h>

// ---------------------------------------------------------------------------
// GCNConv: out = scatter_add(norm * (X W)[src] -> dst) + b
// N=50000, E=600000, D=128.  f32 WMMA GEMM + L2-resident gather/atomic-scatter.
// ---------------------------------------------------------------------------

typedef float v2f __attribute__((ext_vector_type(2)));
typedef float v8f __attribute__((ext_vector_type(8)));

#define D_FEAT 128

// ---------------- degree / normalization -----------------------------------

__global__ void init_deg_kernel(float* __restrict__ deg, int n) {
    int i = blockIdx.x * blockDim.x + threadIdx.x;
    if (i < n) deg[i] = 1.0f;   // self-loop contributes 1 to every node
}

__global__ void edge_deg_kernel(const int* __restrict__ dst,
                                float* __restrict__ deg, int n_edges) {
    int i = blockIdx.x * blockDim.x + threadIdx.x;
    if (i < n_edges) {
        unsafeAtomicAdd(&deg[dst[i]], 1.0f);   // global_atomic_add_f32
    }
}

__global__ void make_dis_kernel(float* __restrict__ deg, int n) {
    int i = blockIdx.x * blockDim.x + threadIdx.x;
    if (i < n) deg[i] = rsqrtf(deg[i]);        // deg >= 1 always
}

// ---------------- H = X * W via v_wmma_f32_16x16x4_f32 ---------------------
// One wave computes a 16x128 strip of H.  Block = 4 waves = 64 rows.
// W (128x128 f32 = 64KB) staged in LDS once per block, K-PAIR interleaved:
//   element (k, col) lives at sW[(k/2)*256 + col*2 + (k&1)]
// so each B fragment (K-pair x column) is ONE ds_load_b64 into an aligned
// VGPR pair -- no v_dual_mov shuffling in front of the WMMAs.

__global__ __launch_bounds__(128) void gemm_wmma_kernel(
        const float* __restrict__ X, const float* __restrict__ W,
        float* __restrict__ H, int n_nodes) {
    __shared__ float sW[D_FEAT * D_FEAT];   // 64 KB (WGP has 320 KB LDS)

    const int tid = threadIdx.x;
    // cooperative interleaved stage: 64 pair-rows x 128 cols = 8192 float2
    {
        v2f* sWp = (v2f*)sW;
        for (int i = tid; i < 64 * D_FEAT; i += 128) {
            int kp  = i >> 7;        // K-pair index 0..63
            int col = i & (D_FEAT - 1);
            v2f p;
            p.x = W[(2 * kp)     * D_FEAT + col];  // coalesced across threads
            p.y = W[(2 * kp + 1) * D_FEAT + col];
            sWp[i] = p;              // ds_store_b64
        }
    }
    __syncthreads();

    const int wave = tid >> 5;
    const int lane = tid & 31;
    const int mbase = (blockIdx.x * 4 + wave) * 16;
    if (mbase >= n_nodes) return;            // whole-wave uniform: EXEC stays full

    const int lrow = lane & 15;              // row (A) / column (B,C,D) within tile
    const int hi   = lane >> 4;              // 0 -> K pair {0,1}, 1 -> K pair {2,3}

    int arow = mbase + lrow;
    if (arow >= n_nodes) arow = n_nodes - 1; // clamp loads; stores are guarded
    const float* xrow = X + (size_t)arow * D_FEAT;

    // pull the rest of this wave's X footprint toward the caches early
    __builtin_prefetch(xrow + 64, 0, 0);     // global_prefetch_b8

    v8f acc[8];
#pragma unroll
    for (int n = 0; n < 8; ++n) acc[n] = (v8f){};

    const v2f* sWp = (const v2f*)sW;
    for (int k0 = 0; k0 < D_FEAT; k0 += 4) {
        // A 16x4 layout: lane half 0 carries K={k0,k0+1}, half 1 K={k0+2,k0+3}
        v2f a = *(const v2f*)(xrow + k0 + 2 * hi);      // global_load_b64
        const v2f* brow = sWp + ((k0 >> 1) + hi) * D_FEAT;
#pragma unroll
        for (int n = 0; n < 8; ++n) {
            v2f b = brow[n * 16 + lrow];                // ds_load_b64
            acc[n] = __builtin_amdgcn_wmma_f32_16x16x4_f32(
                         /*neg_a=*/false, a, /*neg_b=*/false, b,
                         /*c_mod=*/(short)0, acc[n],
                         /*reuse_a=*/false, /*reuse_b=*/false);
        }
    }

    // C/D 16x16 f32 layout: VGPR r holds row M=r (lanes 0-15) / M=r+8 (lanes 16-31)
#pragma unroll
    for (int n = 0; n < 8; ++n) {
#pragma unroll
        for (int r = 0; r < 8; ++r) {
            int row = mbase + r + 8 * hi;
            if (row < n_nodes)
                H[(size_t)row * D_FEAT + n * 16 + lrow] = acc[n][r];
        }
    }
}

// ---------------- out = broadcast(b) ---------------------------------------

__global__ void init_out_kernel(const float* __restrict__ b,
                                float* __restrict__ out, long long total) {
    long long i = (long long)blockIdx.x * blockDim.x + threadIdx.x;
    if (i < total) out[i] = b[i & (D_FEAT - 1)];
}

// ---------------- edge gather + normalize + atomic scatter ------------------
// One wave per logical edge; lane l handles dims [4l, 4l+4).  Gathers are
// 512B/wave coalesced; h and out are L2-resident (25.6MB each vs 192MB L2),
// scatter-adds execute as L2-side global_atomic_add_f32.

__global__ void scatter_kernel(const float* __restrict__ H,
                               const float* __restrict__ dis,
                               const int* __restrict__ src,
                               const int* __restrict__ dst,
                               float* __restrict__ out,
                               int n_edges, int n_nodes) {
    long long gtid = (long long)blockIdx.x * blockDim.x + threadIdx.x;
    long long e    = gtid >> 5;
    int       lane = (int)(gtid & 31);
    long long total = (long long)n_edges + n_nodes;
    if (e >= total) return;

    int s, d;
    if (e < n_edges) { s = src[e]; d = dst[e]; }
    else             { s = (int)(e - n_edges); d = s; }   // self loop

    float nrm = dis[s] * dis[d];
    float4 hv = ((const float4*)(H + (size_t)s * D_FEAT))[lane];

    float* op = out + (size_t)d * D_FEAT + lane * 4;
    unsafeAtomicAdd(op + 0, hv.x * nrm);
    unsafeAtomicAdd(op + 1, hv.y * nrm);
    unsafeAtomicAdd(op + 2, hv.z * nrm);
    unsafeAtomicAdd(op + 3, hv.w * nrm);
}

// ---------------------------------------------------------------------------

extern "C" void kernel_launch(void* const* d_in, const int* in_sizes, int n_in,
                              void* d_out, int out_size, void* d_ws, size_t ws_size,
                              hipStream_t stream) {
    const float* x    = (const float*)d_in[0];   // [N,128]
    const int*   edge = (const int*)  d_in[1];   // [2,E]
    const float* W    = (const float*)d_in[2];   // [128,128]
    const float* b    = (const float*)d_in[3];   // [128]

    const int n_nodes = in_sizes[0] / D_FEAT;
    const int n_edges = in_sizes[1] / 2;
    const int* src = edge;
    const int* dst = edge + n_edges;

    float* out = (float*)d_out;
    float* deg = (float*)d_ws;                   // N floats, becomes dis in place
    float* h   = deg + n_nodes;                  // N*128 floats (~25.6 MB)

    const int T = 256;

    // 1) deg = 1 (self loops)
    init_deg_kernel<<<(n_nodes + T - 1) / T, T, 0, stream>>>(deg, n_nodes);
    // 2) deg[dst] += 1 over edges
    edge_deg_kernel<<<(n_edges + T - 1) / T, T, 0, stream>>>(dst, deg, n_edges);
    // 3) dis = rsqrt(deg)
    make_dis_kernel<<<(n_nodes + T - 1) / T, T, 0, stream>>>(deg, n_nodes);
    // 4) h = x @ W  (WMMA f32)
    {
        int blocks = (n_nodes + 63) / 64;        // 4 waves * 16 rows per block
        gemm_wmma_kernel<<<blocks, 128, 0, stream>>>(x, W, h, n_nodes);
    }
    // 5) out = b broadcast
    {
        long long total = (long long)n_nodes * D_FEAT;
        init_out_kernel<<<(int)((total + T - 1) / T), T, 0, stream>>>(b, out, total);
    }
    // 6) out[dst] += h[src] * dis[src] * dis[dst]  (incl. self loops)
    {
        long long lanes = ((long long)n_edges + n_nodes) * 32;
        scatter_kernel<<<(int)((lanes + T - 1) / T), T, 0, stream>>>(
            h, deg, src, dst, out, n_edges, n_nodes);
    }
}